// RegressionMatcher_74680891342868
// MI455X (gfx1250) — compile-verified
//
#include <hip/hip_runtime.h>
#include <hip/hip_bf16.h>
#include <math.h>

// Problem sizes (fixed by reference setup_inputs)
#define N_PTS 8192
#define IMG_H 504
#define IMG_W 1008
#define SAMPLE_THRESH 0.05f

typedef __attribute__((ext_vector_type(2))) float v2f;
typedef __attribute__((ext_vector_type(8))) float v8f;

// Fast HW sqrt (single v_sqrt_f32). Monotone; used identically for per-element
// D and for the row/col minima, so the exact-equality mask stays consistent.
__device__ __forceinline__ float hw_sqrt(float x) {
    return __builtin_amdgcn_sqrtf(x);
}

// Raw v_max_num_f32 without clang's sNaN-quieting canonicalization (WMMA
// output is never a signaling NaN). One instruction instead of two.
__device__ __forceinline__ float clamp_lo(float x, float lo) {
    float r;
    asm("v_max_num_f32 %0, %1, %2" : "=v"(r) : "v"(x), "v"(lo));
    return r;
}

__device__ __forceinline__ unsigned umin2(unsigned a, unsigned b) {
    return a < b ? a : b;
}

// ---------------------------------------------------------------------------
// Bilinear grid-sample (zeros padding, align_corners=False) of warp channels
// 2,3 and certainty at x_A points; build augmented A rows:
//   A4[i] = (-2*ax, -2*ay, ax^2+ay^2, 1)
// so that A4[i] . B4[j] = ||a||^2 + ||b||^2 - 2 a.b = D^2  with K=4 (WMMA 16x16x4).
// ---------------------------------------------------------------------------
__device__ __forceinline__ float fetch_warp_c(const float* __restrict__ warp,
                                              int y, int x, int c) {
    if (x < 0 || x >= IMG_W || y < 0 || y >= IMG_H) return 0.0f;
    return warp[((size_t)y * IMG_W + x) * 4 + c];
}
__device__ __forceinline__ float fetch_cert(const float* __restrict__ cert,
                                            int y, int x) {
    if (x < 0 || x >= IMG_W || y < 0 || y >= IMG_H) return 0.0f;
    return cert[(size_t)y * IMG_W + x];
}

__global__ void rm_sample_A(const float* __restrict__ x_A,
                            const float* __restrict__ warp,
                            const float* __restrict__ certainty,
                            float4* __restrict__ A4,
                            float* __restrict__ certA) {
    int i = blockIdx.x * blockDim.x + threadIdx.x;
    if (i >= N_PTS) return;
    float px = x_A[2 * i + 0];
    float py = x_A[2 * i + 1];
    float fx = (px + 1.0f) * (IMG_W * 0.5f) - 0.5f;
    float fy = (py + 1.0f) * (IMG_H * 0.5f) - 0.5f;
    float x0f = floorf(fx), y0f = floorf(fy);
    int x0 = (int)x0f, y0 = (int)y0f;
    float wx1 = fx - x0f, wy1 = fy - y0f;
    float wx0 = 1.0f - wx1, wy0 = 1.0f - wy1;
    float w00 = wx0 * wy0, w10 = wx1 * wy0, w01 = wx0 * wy1, w11 = wx1 * wy1;

    float ax = w00 * fetch_warp_c(warp, y0, x0, 2)
             + w10 * fetch_warp_c(warp, y0, x0 + 1, 2)
             + w01 * fetch_warp_c(warp, y0 + 1, x0, 2)
             + w11 * fetch_warp_c(warp, y0 + 1, x0 + 1, 2);
    float ay = w00 * fetch_warp_c(warp, y0, x0, 3)
             + w10 * fetch_warp_c(warp, y0, x0 + 1, 3)
             + w01 * fetch_warp_c(warp, y0 + 1, x0, 3)
             + w11 * fetch_warp_c(warp, y0 + 1, x0 + 1, 3);
    float cv = w00 * fetch_cert(certainty, y0, x0)
             + w10 * fetch_cert(certainty, y0, x0 + 1)
             + w01 * fetch_cert(certainty, y0 + 1, x0)
             + w11 * fetch_cert(certainty, y0 + 1, x0 + 1);

    A4[i] = make_float4(-2.0f * ax, -2.0f * ay, ax * ax + ay * ay, 1.0f);
    certA[i] = cv;
}

__global__ void rm_build_B(const float* __restrict__ x_B,
                           float4* __restrict__ B4) {
    int j = blockIdx.x * blockDim.x + threadIdx.x;
    if (j >= N_PTS) return;
    float bx = x_B[2 * j + 0];
    float by = x_B[2 * j + 1];
    B4[j] = make_float4(bx, by, 1.0f, bx * bx + by * by);
}

__global__ void rm_init_mins(unsigned* __restrict__ rowmin,
                             unsigned* __restrict__ colmin) {
    int i = blockIdx.x * blockDim.x + threadIdx.x;
    if (i < N_PTS) {
        rowmin[i] = 0x7F800000u;  // +inf bits; uint order == float order for vals > 0
        colmin[i] = 0x7F800000u;
    }
}

// Convert reduced min-D^2 (clamped, > 0) into min-D with the same hw sqrt
// used by pass 2, so bitwise equality against per-element D holds.
__global__ void rm_finalize_mins(unsigned* __restrict__ rowmin,
                                 unsigned* __restrict__ colmin) {
    int i = blockIdx.x * blockDim.x + threadIdx.x;
    if (i < N_PTS) {
        rowmin[i] = __float_as_uint(hw_sqrt(__uint_as_float(rowmin[i])));
        colmin[i] = __float_as_uint(hw_sqrt(__uint_as_float(colmin[i])));
    }
}

// ---------------------------------------------------------------------------
// WMMA helper: one 16x16 D^2 tile from augmented rows.
// A frag (16x4 f32): lanes 0-15 hold K=0,1 in v0,v1; lanes 16-31 hold K=2,3.
// B frag (4x16 f32): same split transposed (lanes 0-15: K=0,1; 16-31: K=2,3).
// C/D (16x16 f32):   vgpr r = row r (lanes 0-15) / row r+8 (lanes 16-31), col = lane%16.
// ---------------------------------------------------------------------------
__device__ __forceinline__ v8f d2_tile_wmma(float4 arow, float4 bcol, int grp) {
    v2f a, b;
    a.x = grp ? arow.z : arow.x;
    a.y = grp ? arow.w : arow.y;
    b.x = grp ? bcol.z : bcol.x;
    b.y = grp ? bcol.w : bcol.y;
    v8f c = {};
    c = __builtin_amdgcn_wmma_f32_16x16x4_f32(false, a, false, b,
                                              (short)0, c, false, false);
    return c;
}

// ---------------------------------------------------------------------------
// Pass 1: per-block 128x128 region; reduce CLAMPED D^2 (no sqrt needed: sqrt
// and max are monotone, so min commutes with them). After the clamp all
// values are positive floats, so the min-reductions run in the unsigned-int
// domain (v_min_u32 / v_min3_u32, no NaN canonicalization). Row minima in
// registers (shuffle-reduced), col minima across waves via LDS.
// ---------------------------------------------------------------------------
__global__ void rm_pass1(const float4* __restrict__ A4,
                         const float4* __restrict__ B4,
                         unsigned* __restrict__ rowmin,
                         unsigned* __restrict__ colmin) {
    __shared__ unsigned s_colmin[128];
    const int tid  = threadIdx.x;
    const int wave = tid >> 5;
    const int lane = tid & 31;
    const int grp  = lane >> 4;   // half-wave: which K pair
    const int idx  = lane & 15;   // row (A) / col (B) within tile

    if (tid < 128) s_colmin[tid] = 0x7F800000u;
    __syncthreads();

    const int m_base = blockIdx.y * 128 + wave * 16;
    const int n_blk  = blockIdx.x * 128;

    const float4 arow = A4[m_base + idx];
    unsigned rminU[8];
#pragma unroll
    for (int r = 0; r < 8; ++r) rminU[r] = 0x7F800000u;

#pragma unroll
    for (int t = 0; t < 8; ++t) {
        const int n_base = n_blk + t * 16;
        const float4 bcol = B4[n_base + idx];
        v8f c = d2_tile_wmma(arow, bcol, grp);

        unsigned cminU = 0x7F800000u;
#pragma unroll
        for (int r = 0; r < 8; ++r) {
            unsigned e = __float_as_uint(clamp_lo(c[r], 1e-12f));
            rminU[r] = umin2(rminU[r], e);
            cminU = umin2(cminU, e);
        }
        // combine the two half-wave row groups for this column
        cminU = umin2(cminU, (unsigned)__shfl_xor((int)cminU, 16, 32));
        if (grp == 0)
            atomicMin(&s_colmin[t * 16 + idx], cminU);
    }

    // reduce row minima across the 16 lanes of each half-wave group
#pragma unroll
    for (int r = 0; r < 8; ++r) {
        unsigned v = rminU[r];
        v = umin2(v, (unsigned)__shfl_xor((int)v, 1, 16));
        v = umin2(v, (unsigned)__shfl_xor((int)v, 2, 16));
        v = umin2(v, (unsigned)__shfl_xor((int)v, 4, 16));
        v = umin2(v, (unsigned)__shfl_xor((int)v, 8, 16));
        rminU[r] = v;
    }
    if (idx == 0) {
#pragma unroll
        for (int r = 0; r < 8; ++r)
            atomicMin(&rowmin[m_base + grp * 8 + r], rminU[r]);
    }

    __syncthreads();
    if (tid < 128)
        atomicMin(&colmin[n_blk + tid], s_colmin[tid]);
}

// ---------------------------------------------------------------------------
// Pass 2: recompute D^2 (bitwise identical WMMA), hw-sqrt to D, apply
// mutual-min + certainty mask, stage 16x128 tile in LDS, write coalesced
// float4 rows (the only full pass over the 256 MB output -> bandwidth-optimal).
// ---------------------------------------------------------------------------
__global__ void rm_pass2(const float4* __restrict__ A4,
                         const float4* __restrict__ B4,
                         const unsigned* __restrict__ rowminD,
                         const unsigned* __restrict__ colminD,
                         const float* __restrict__ certA,
                         float* __restrict__ out) {
    __shared__ float s_tile[16 * 128];
    const int tid  = threadIdx.x;
    const int wave = tid >> 5;
    const int lane = tid & 31;
    const int grp  = lane >> 4;
    const int idx  = lane & 15;

    const int m_base = blockIdx.y * 16;
    const int n_blk  = blockIdx.x * 128;
    const int n_base = n_blk + wave * 16;

    const float4 arow = A4[m_base + idx];
    const float4 bcol = B4[n_base + idx];
    v8f c = d2_tile_wmma(arow, bcol, grp);

    const unsigned cmnU = colminD[n_base + idx];
#pragma unroll
    for (int r = 0; r < 8; ++r) {
        const int m = m_base + grp * 8 + r;
        float d = hw_sqrt(clamp_lo(c[r], 1e-12f));
        unsigned dU = __float_as_uint(d);
        bool ok = (dU == rowminD[m]) && (dU == cmnU) &&
                  (certA[m] > SAMPLE_THRESH);
        s_tile[(grp * 8 + r) * 128 + wave * 16 + idx] = ok ? d : 0.0f;
    }
    __syncthreads();

    // 16 rows x 128 cols = 512 float4; 256 threads -> 2 stores each,
    // 128 consecutive floats per 32-lane group (fully coalesced).
    const float4* s4 = (const float4*)s_tile;
    float4* o4 = (float4*)out;
#pragma unroll
    for (int i = tid; i < 512; i += 256) {
        const int row = i >> 5;
        const int c4  = i & 31;
        o4[(size_t)(m_base + row) * (N_PTS / 4) + (n_blk / 4) + c4] = s4[i];
    }
}

// ---------------------------------------------------------------------------
extern "C" void kernel_launch(void* const* d_in, const int* in_sizes, int n_in,
                              void* d_out, int out_size, void* d_ws, size_t ws_size,
                              hipStream_t stream) {
    const float* x_A       = (const float*)d_in[0];
    const float* x_B       = (const float*)d_in[1];
    const float* warp      = (const float*)d_in[2];
    const float* certainty = (const float*)d_in[3];
    float* out = (float*)d_out;

    char* ws = (char*)d_ws;
    float4*   A4     = (float4*)(ws);
    float4*   B4     = (float4*)(ws + (size_t)N_PTS * 16);
    float*    certA  = (float*)(ws + (size_t)N_PTS * 32);
    unsigned* rowmin = (unsigned*)(ws + (size_t)N_PTS * 36);
    unsigned* colmin = (unsigned*)(ws + (size_t)N_PTS * 40);

    rm_sample_A<<<N_PTS / 256, 256, 0, stream>>>(x_A, warp, certainty, A4, certA);
    rm_build_B<<<N_PTS / 256, 256, 0, stream>>>(x_B, B4);
    rm_init_mins<<<N_PTS / 256, 256, 0, stream>>>(rowmin, colmin);

    rm_pass1<<<dim3(N_PTS / 128, N_PTS / 128), 256, 0, stream>>>(A4, B4, rowmin, colmin);
    rm_finalize_mins<<<N_PTS / 256, 256, 0, stream>>>(rowmin, colmin);
    rm_pass2<<<dim3(N_PTS / 128, N_PTS / 16), 256, 0, stream>>>(A4, B4, rowmin, colmin,
                                                                certA, out);
}